// GraphModule_41575283426067
// MI455X (gfx1250) — compile-verified
//
#include <hip/hip_runtime.h>
#include <hip/hip_bf16.h>
#include <math.h>

// ---------------- problem constants (match reference module) ----------------
#define N_NODES 8192
#define N_EDGES 131072
#define HDIM    256
#define NHEADS  4
#define HD      64          // head dim
#define SZ      4096        // number of user (query) nodes
#define MASKW   (N_NODES/32)   // 256 u32 words per mask row

typedef __attribute__((ext_vector_type(2))) float v2f;
typedef __attribute__((ext_vector_type(8))) float v8f;

// fp32 WMMA: D(16x16,f32) = A(16x4,f32) * B(4x16,f32) + C
// 8-arg form: (neg_a, A, neg_b, B, c_mod, C, reuse_a, reuse_b)
__device__ __forceinline__ v8f wmma_f32(v2f a, v2f b, v8f c) {
    return __builtin_amdgcn_wmma_f32_16x16x4_f32(false, a, false, b,
                                                 (short)0, c, false, false);
}

// ---------------------------------------------------------------------------
// WMMA GEMM: C[M x Nout] = A[M x K] @ W[Nout x K]^T (+ bias[Nout])
// One 16x64 output tile per wave (4 accumulators); software-pipelined K-loop:
// fragments for step k+1 are loaded before the WMMAs of step k are issued, so
// the 4 WMMAs of each step overlap with the 5 in-flight loads of the next.
// A fragment (16x4):  lane<16 -> row m0+lane,    K = k0+{0,1}
//                     lane>=16 -> row m0+lane-16, K = k0+{2,3}
// B fragment (4x16):  lane<16 -> col n+lane (= W row), K = k0+{0,1}
//                     lane>=16 -> col n+lane-16,        K = k0+{2,3}
// Requires M % 16 == 0, Nout % 64 == 0, K % 4 == 0, K >= 8.
// ---------------------------------------------------------------------------
__global__ __launch_bounds__(256)
void k_gemm_wmma(const float* __restrict__ A, const float* __restrict__ W,
                 const float* __restrict__ bias, float* __restrict__ C,
                 int M, int K, int Nout, int lda, int ldw, int ldc) {
    const int lane = threadIdx.x & 31;
    const int wv   = threadIdx.x >> 5;
    const int ntn  = Nout >> 6;                    // 64-wide column tiles
    const int tile = blockIdx.x * (blockDim.x >> 5) + wv;
    if (tile >= (M >> 4) * ntn) return;            // wave-uniform (EXEC stays full)
    const int m0  = (tile / ntn) << 4;
    const int n0c = (tile % ntn) << 6;
    const int lr = lane & 15, hi = lane >> 4;

    const float* arow  = A + (size_t)(m0 + lr) * lda + 2 * hi;
    const float* bbase = W + (size_t)(n0c + lr) * ldw + 2 * hi;
    const size_t bstep = (size_t)16 * ldw;         // 16 W-rows per sub-tile

    v8f acc[4];
#pragma unroll
    for (int t = 0; t < 4; ++t) acc[t] = (v8f){0.f,0.f,0.f,0.f,0.f,0.f,0.f,0.f};

    // prologue: fragments for k0 = 0
    v2f a_cur = *(const v2f*)(arow);
    v2f b_cur[4];
#pragma unroll
    for (int t = 0; t < 4; ++t) b_cur[t] = *(const v2f*)(bbase + (size_t)t * bstep);

    for (int k0 = 0; k0 < K - 4; k0 += 4) {
        const int k1 = k0 + 4;
        __builtin_prefetch(arow + k1 + 64, 0, 1);  // -> global_prefetch_b8
        // issue next step's loads before consuming current fragments
        v2f a_nxt = *(const v2f*)(arow + k1);
        v2f b_nxt[4];
#pragma unroll
        for (int t = 0; t < 4; ++t)
            b_nxt[t] = *(const v2f*)(bbase + (size_t)t * bstep + k1);
#pragma unroll
        for (int t = 0; t < 4; ++t) acc[t] = wmma_f32(a_cur, b_cur[t], acc[t]);
        a_cur = a_nxt;
#pragma unroll
        for (int t = 0; t < 4; ++t) b_cur[t] = b_nxt[t];
    }
    // epilogue: last K-step
#pragma unroll
    for (int t = 0; t < 4; ++t) acc[t] = wmma_f32(a_cur, b_cur[t], acc[t]);

#pragma unroll
    for (int t = 0; t < 4; ++t) {
        const int n0 = n0c + 16 * t;
        const float bv = bias ? bias[n0 + lr] : 0.0f;
        float* out = C + n0 + lr;
#pragma unroll
        for (int r = 0; r < 8; ++r)
            out[(size_t)(m0 + r + 8 * hi) * ldc] = acc[t][r] + bv;
    }
}

// ---------------------------- GCN scatter path -----------------------------
__global__ void k_deg_init(float* deg) {
    int i = blockIdx.x * blockDim.x + threadIdx.x;
    if (i < N_NODES) deg[i] = 1.0f;                 // self loop
}
__global__ void k_edge_deg(const int* __restrict__ dst, float* deg) {
    int e = blockIdx.x * blockDim.x + threadIdx.x;
    if (e < N_EDGES) atomicAdd(&deg[dst[e]], 1.0f);
}
__global__ void k_dinv(const float* __restrict__ deg, float* dinv) {
    int i = blockIdx.x * blockDim.x + threadIdx.x;
    if (i < N_NODES) dinv[i] = rsqrtf(deg[i]);
}
__global__ void k_res_init(const float* __restrict__ x, const float* __restrict__ dinv,
                           const float* __restrict__ bg, float* __restrict__ res) {
    int i = blockIdx.x * blockDim.x + threadIdx.x;
    if (i < N_NODES * HDIM) {
        int node = i >> 8, h = i & 255;
        float dv = dinv[node];
        res[i] = dv * dv * x[i] + bg[h];            // self-loop term + bias
    }
}
__global__ void k_edge_scatter(const int* __restrict__ src, const int* __restrict__ dst,
                               const float* __restrict__ dinv, const float* __restrict__ x,
                               float* __restrict__ res) {
    int t = blockIdx.x * blockDim.x + threadIdx.x;  // E * 64 threads, 4 feats each
    int e = t >> 6;
    if (e < N_EDGES) {
        int h = (t & 63) << 2;
        int s = src[e], d = dst[e];
        float nm = dinv[s] * dinv[d];
        const float4 xv = *(const float4*)(x + (size_t)s * HDIM + h);
        float* o = res + (size_t)d * HDIM + h;
        atomicAdd(o + 0, nm * xv.x);
        atomicAdd(o + 1, nm * xv.y);
        atomicAdd(o + 2, nm * xv.z);
        atomicAdd(o + 3, nm * xv.w);
    }
}

// ------------------------------ neighbor mask ------------------------------
__global__ void k_mask_init(unsigned* mw) {
    int i = blockIdx.x * blockDim.x + threadIdx.x;
    if (i < SZ * MASKW) {
        int row = i / MASKW, w = i % MASKW;
        mw[i] = ((row >> 5) == w) ? (1u << (row & 31)) : 0u;   // diagonal
    }
}
__global__ void k_mask_edges(const int* __restrict__ src, const int* __restrict__ dst,
                             unsigned* mw) {
    int e = blockIdx.x * blockDim.x + threadIdx.x;
    if (e < N_EDGES) {
        int s = src[e], d = dst[e];
        if (d < SZ) atomicOr(&mw[(size_t)d * MASKW + (s >> 5)], 1u << (s & 31));
        if (s < SZ) atomicOr(&mw[(size_t)s * MASKW + (d >> 5)], 1u << (d & 31));
    }
}

// ---------------------------------------------------------------------------
// Masked flash attention. qkv layout: [node][768] = [q(0:256) | k(256:512) | v(512:768)],
// head-major within each 256 (head*64 + d). One wave per (head, 16-query block).
// Row softmax stats kept in C-tile layout: stat[r] valid for row r (+8 on hi half).
// ---------------------------------------------------------------------------
__global__ __launch_bounds__(128)
void k_attn(const float* __restrict__ qkv, const unsigned* __restrict__ mw,
            float* __restrict__ ctx) {
    __shared__ __align__(16) float ldsP[4][256];    // 16x16 P tile per wave
    const int lane = threadIdx.x & 31;
    const int wv   = threadIdx.x >> 5;
    const int task = blockIdx.x * 4 + wv;           // 1024 tasks exactly
    const int head = task & (NHEADS - 1);
    const int q0   = (task >> 2) << 4;
    const int lr = lane & 15, hi = lane >> 4;
    const int h0 = head * HD;

    // Q fragments for all of head-dim (16 chunks of K=4), kept in registers.
    const float* qb = qkv + (size_t)(q0 + lr) * 768 + h0 + 2 * hi;
    v2f aq[16];
#pragma unroll
    for (int kk = 0; kk < 16; ++kk) aq[kk] = *(const v2f*)(qb + 4 * kk);

    v8f acc[4];
#pragma unroll
    for (int t = 0; t < 4; ++t) acc[t] = (v8f){0.f,0.f,0.f,0.f,0.f,0.f,0.f,0.f};
    float mrow[8], lrow[8];
#pragma unroll
    for (int r = 0; r < 8; ++r) { mrow[r] = -1e30f; lrow[r] = 0.0f; }

    float* lp = ldsP[wv];

    for (int n0 = 0; n0 < N_NODES; n0 += 16) {
        // ---- S tile = Q[16 x 64] * K^T[64 x 16] via 16 fp32 WMMAs ----
        // Double-buffered K fragments: load kk+1 before issuing WMMA kk.
        v8f s = {0.f,0.f,0.f,0.f,0.f,0.f,0.f,0.f};
        const float* kb = qkv + (size_t)(n0 + lr) * 768 + 256 + h0 + 2 * hi;
        v2f b_cur = *(const v2f*)(kb);
#pragma unroll
        for (int kk = 0; kk < 16; ++kk) {
            v2f b_nxt;
            if (kk < 15) b_nxt = *(const v2f*)(kb + 4 * (kk + 1));
            s = wmma_f32(aq[kk], b_cur, s);
            if (kk < 15) b_cur = b_nxt;
        }

        // ---- mask + online softmax (stats in C-tile layout) ----
        const int w0 = n0 >> 5;
        const int bitofs = (n0 & 16) + lr;
        float sc[8];
#pragma unroll
        for (int r = 0; r < 8; ++r) {
            unsigned wb = mw[(size_t)(q0 + r + 8 * hi) * MASKW + w0];
            bool on = (wb >> bitofs) & 1u;
            float sv = on ? s[r] * 0.125f : -1e30f;          // 1/sqrt(64)
            float t = sv;                                     // row max in half
            t = fmaxf(t, __shfl_xor(t, 1, 32));
            t = fmaxf(t, __shfl_xor(t, 2, 32));
            t = fmaxf(t, __shfl_xor(t, 4, 32));
            t = fmaxf(t, __shfl_xor(t, 8, 32));
            float nm = fmaxf(mrow[r], t);
            sc[r] = __expf(mrow[r] - nm);
            mrow[r] = nm;
            float pv = on ? __expf(sv - nm) : 0.0f;
            float ts = pv;                                    // row sum in half
            ts += __shfl_xor(ts, 1, 32);
            ts += __shfl_xor(ts, 2, 32);
            ts += __shfl_xor(ts, 4, 32);
            ts += __shfl_xor(ts, 8, 32);
            lrow[r] = lrow[r] * sc[r] + ts;
            lp[(r + 8 * hi) * 16 + lr] = pv;                  // stage P tile in LDS
        }
#pragma unroll
        for (int r = 0; r < 8; ++r) {                         // rescale ctx accum
            acc[0][r] *= sc[r]; acc[1][r] *= sc[r];
            acc[2][r] *= sc[r]; acc[3][r] *= sc[r];
        }

        // ---- ctx += P[16 x 16] * V[16 x 64] via 16 fp32 WMMAs ----
#pragma unroll
        for (int k4 = 0; k4 < 4; ++k4) {
            const int k0 = k4 * 4;
            // P re-read from LDS in A-fragment layout
            v2f ap = *(const v2f*)&lp[lr * 16 + k0 + 2 * hi];
            const float* vb = qkv + (size_t)(n0 + k0 + 2 * hi) * 768 + 512 + h0;
            // issue all 8 V loads for this K-chunk before the 4 WMMAs
            v2f bf[4];
#pragma unroll
            for (int dt = 0; dt < 4; ++dt) {
                bf[dt].x = vb[dt * 16 + lr];
                bf[dt].y = vb[768 + dt * 16 + lr];
            }
#pragma unroll
            for (int dt = 0; dt < 4; ++dt)
                acc[dt] = wmma_f32(ap, bf[dt], acc[dt]);
        }
    }

    // ---- normalize + write ctx [SZ x HDIM] (head-major columns) ----
#pragma unroll
    for (int r = 0; r < 8; ++r) {
        float inv = 1.0f / lrow[r];
#pragma unroll
        for (int dt = 0; dt < 4; ++dt)
            ctx[(size_t)(q0 + r + 8 * hi) * HDIM + h0 + dt * 16 + lr] = acc[dt][r] * inv;
    }
}

__global__ void k_copy(const float* __restrict__ in, float* __restrict__ out, int n) {
    int i = blockIdx.x * blockDim.x + threadIdx.x;
    if (i < n) out[i] = in[i];
}

// ---------------------------------------------------------------------------
extern "C" void kernel_launch(void* const* d_in, const int* in_sizes, int n_in,
                              void* d_out, int out_size, void* d_ws, size_t ws_size,
                              hipStream_t stream) {
    const float* rep  = (const float*)d_in[0];
    const int*   eidx = (const int*)d_in[1];     // [2, E]
    // d_in[2] edge_type unused by reference; d_in[3] size == SZ (fixed)
    const float* Wg  = (const float*)d_in[4];
    const float* bg  = (const float*)d_in[5];
    const float* ipw = (const float*)d_in[6];
    const float* ipb = (const float*)d_in[7];
    const float* ow  = (const float*)d_in[8];
    const float* ob  = (const float*)d_in[9];
    float* out = (float*)d_out;

    // workspace layout (~51 MB)
    char* p = (char*)d_ws;
    float* x    = (float*)p;  p += (size_t)N_NODES * HDIM * 4;
    float* res  = (float*)p;  p += (size_t)N_NODES * HDIM * 4;
    float* qkv  = (float*)p;  p += (size_t)N_NODES * 768  * 4;
    float* ctx  = (float*)p;  p += (size_t)SZ * HDIM * 4;
    float* deg  = (float*)p;  p += (size_t)N_NODES * 4;
    float* dinv = (float*)p;  p += (size_t)N_NODES * 4;
    unsigned* mwv = (unsigned*)p;

    const int* esrc = eidx;              // edge_index[0]
    const int* edst = eidx + N_EDGES;    // edge_index[1]

    // --- GCN: x = rep @ Wg^T ---
    k_deg_init<<<(N_NODES + 255) / 256, 256, 0, stream>>>(deg);
    k_mask_init<<<(SZ * MASKW + 255) / 256, 256, 0, stream>>>(mwv);
    {
        int tiles = (N_NODES / 16) * (HDIM / 64);   // 2048 wave-tasks
        k_gemm_wmma<<<tiles / 8, 256, 0, stream>>>(rep, Wg, nullptr, x,
                                                   N_NODES, HDIM, HDIM,
                                                   HDIM, HDIM, HDIM);
    }
    k_edge_deg<<<(N_EDGES + 255) / 256, 256, 0, stream>>>(edst, deg);
    k_dinv<<<(N_NODES + 255) / 256, 256, 0, stream>>>(deg, dinv);
    k_res_init<<<(N_NODES * HDIM + 255) / 256, 256, 0, stream>>>(x, dinv, bg, res);
    k_edge_scatter<<<(N_EDGES * 64 + 255) / 256, 256, 0, stream>>>(esrc, edst, dinv, x, res);
    k_mask_edges<<<(N_EDGES + 255) / 256, 256, 0, stream>>>(esrc, edst, mwv);

    // --- fused QKV projection: qkv = res @ in_proj_w^T + in_proj_b  [N x 768] ---
    {
        int tiles = (N_NODES / 16) * (768 / 64);    // 6144 wave-tasks
        k_gemm_wmma<<<tiles / 8, 256, 0, stream>>>(res, ipw, ipb, qkv,
                                                   N_NODES, HDIM, 768,
                                                   HDIM, HDIM, 768);
    }

    // --- masked flash attention -> ctx [SZ x HDIM] ---
    k_attn<<<(NHEADS * (SZ / 16)) / 4, 128, 0, stream>>>(qkv, mwv, ctx);

    // --- output projection: users = ctx @ ow^T + ob -> out[0 : SZ*HDIM] ---
    {
        int tiles = (SZ / 16) * (HDIM / 64);        // 1024 wave-tasks
        k_gemm_wmma<<<tiles / 8, 256, 0, stream>>>(ctx, ow, ob, out,
                                                   SZ, HDIM, HDIM,
                                                   HDIM, HDIM, HDIM);
    }

    // --- out[SZ*HDIM :] = res[SZ*HDIM :] ---
    k_copy<<<(SZ * HDIM + 255) / 256, 256, 0, stream>>>(res + (size_t)SZ * HDIM,
                                                        out + (size_t)SZ * HDIM,
                                                        SZ * HDIM);
}